// OuterConcatenate_74474732913011
// MI455X (gfx1250) — compile-verified
//
#include <hip/hip_runtime.h>
#include <math.h>

typedef __attribute__((ext_vector_type(2))) float v2f;
typedef __attribute__((ext_vector_type(4))) float v4f;
typedef __attribute__((ext_vector_type(8))) float v8f;

#define B_    2
#define L_    512
#define NIN   64
#define NOUT  96
#define KK    192          // 3 taps x 64 channels, ordered k = dy*64 + c
#define EPSV  1e-5f

// ---------------------------------------------------------------------------
// Fold conv weights into edge-class collapsed matrices, pair-interleaved for
// b64 B-fragment loads:  W2[e][k>>1][o][k&1],  k = d*64 + c  (d = dy or dx).
// Wr: sum over dx in valid(e) of W[o][c][dy=d][dx]            (c < 64)
// Wc: sum over dy in valid(e) of W[o][64+c][dy][dx=d]         (c < 64)
// valid(0) = {1,2}  (first row/col), valid(1) = all, valid(2) = {0,1} (last)
// ---------------------------------------------------------------------------
__global__ void prep_weights(const float* __restrict__ W,
                             float* __restrict__ Wr, float* __restrict__ Wc) {
    int idx = blockIdx.x * blockDim.x + threadIdx.x;       // over 3*192*96
    if (idx >= 3 * KK * NOUT) return;
    int o = idx % NOUT;
    int k = (idx / NOUT) % KK;
    int e = idx / (NOUT * KK);
    int c = k & 63;
    int d = k >> 6;                                        // dy for Wr, dx for Wc
    float sr = 0.f, sc = 0.f;
    for (int t = 0; t < 3; ++t) {
        bool ok = (e == 1) || (e == 0 && t >= 1) || (e == 2 && t <= 1);
        if (ok) {
            sr += W[((o * 128 + c) * 3 + d) * 3 + t];          // W[o][c][dy=d][dx=t]
            sc += W[((o * 128 + 64 + c) * 3 + t) * 3 + d];     // W[o][64+c][dy=t][dx=d]
        }
    }
    int dst = ((e * (KK / 2) + (k >> 1)) * NOUT + o) * 2 + (k & 1);
    Wr[dst] = sr;
    Wc[dst] = sc;
}

// ---------------------------------------------------------------------------
// WMMA fp32 GEMM: Out[b][e][i][o] = sum_k A(b,i,k) * Wm[e][k][o]
// A(b,i,k) = X[b, i + (k>>6) - 1, k&63] (zero outside [0,512)) -- virtual.
// One wave32 per 16x16 tile, K stepped by 4 with V_WMMA_F32_16X16X4_F32.
// dy-major K order: per dy-segment the input row is fixed, so A fragments are
// contiguous b64 loads; boundary handled by clamped address + value select
// (no EXEC divergence). grid = (32 Mtiles, 6 Ntiles, b*3+e).
// ---------------------------------------------------------------------------
__global__ __launch_bounds__(32)
void gemm_rc(const float* __restrict__ X,     // [B][512][64]
             const float* __restrict__ Wm,    // [3][96][96][2] pair-interleaved
             float* __restrict__ Out) {       // [B][3][512][96]
    const int b    = blockIdx.z / 3;
    const int e    = blockIdx.z % 3;
    const int lane = threadIdx.x;
    const int half = lane >> 4;
    const int l    = lane & 15;
    const int Mb   = blockIdx.x * 16;
    const int Nb   = blockIdx.y * 16;
    const float* Wme = Wm + (size_t)e * ((KK / 2) * NOUT * 2);
    const float* Xb  = X + (size_t)b * (L_ * NIN);
    const int n = Nb + l;

    v8f acc = {};
    for (int dy = 0; dy < 3; ++dy) {
        const int row = Mb + l + dy - 1;                   // per-lane input row
        const bool ok = (row >= 0) && (row < L_);
        const int rc  = row < 0 ? 0 : (row > L_ - 1 ? L_ - 1 : row);
        const float* Xrow = Xb + rc * NIN;
        for (int c0 = 0; c0 < NIN; c0 += 4) {
            const int ca = c0 + 2 * half;                  // this half: K=ca, ca+1
            // A fragment: contiguous pair, zeroed by select at grid edges
            v2f a = *(const v2f*)(Xrow + ca);
            a.x = ok ? a.x : 0.f;
            a.y = ok ? a.y : 0.f;
            // B fragment: contiguous pair from pair-interleaved weights
            const int kp = dy * (NIN / 2) + (c0 >> 1) + half;
            v2f bv = *(const v2f*)(Wme + ((size_t)kp * NOUT + n) * 2);
            acc = __builtin_amdgcn_wmma_f32_16x16x4_f32(
                      false, a, false, bv, (short)0, acc, false, false);
        }
    }
    // D layout: VGPR r -> M = r (lanes 0-15) / M = 8+r (lanes 16-31), N = lane&15
    float* Ob = Out + (((size_t)b * 3 + e) * L_ + Mb) * NOUT + Nb;
    for (int r = 0; r < 8; ++r) {
        const int rowD = r + 8 * half;
        Ob[rowD * NOUT + l] = acc[r];
    }
}

// ---------------------------------------------------------------------------
// Analytic InstanceNorm statistics. Since out(i,j) = R(i,jc(j)) + C(j,ic(i)):
//   Sum   = S_R[0] + 510*S_R[1] + S_R[2] + S_C[0] + 510*S_C[1] + S_C[2]
//   SumSq = Q_R + Q_C + 2*Cross    (Cross = 9-block factorized product)
// One block per (b,o), 256 threads reduce over the 512 rows/cols.
// ---------------------------------------------------------------------------
__global__ __launch_bounds__(256)
void stats(const float* __restrict__ R, const float* __restrict__ C,
           float* __restrict__ meanv, float* __restrict__ invv) {
    const int bo = blockIdx.x;
    const int b = bo / NOUT, o = bo % NOUT;
    const int t = threadIdx.x;

    float sR[3] = {0, 0, 0}, sC[3] = {0, 0, 0}, qR = 0.f, qC = 0.f;
    for (int i = t; i < L_; i += 256) {
        float r0 = R[((b * 3 + 0) * L_ + i) * NOUT + o];
        float r1 = R[((b * 3 + 1) * L_ + i) * NOUT + o];
        float r2 = R[((b * 3 + 2) * L_ + i) * NOUT + o];
        sR[0] += r0; sR[1] += r1; sR[2] += r2;
        qR += 510.f * r1 * r1 + r0 * r0 + r2 * r2;
        float c0 = C[((b * 3 + 0) * L_ + i) * NOUT + o];
        float c1 = C[((b * 3 + 1) * L_ + i) * NOUT + o];
        float c2 = C[((b * 3 + 2) * L_ + i) * NOUT + o];
        sC[0] += c0; sC[1] += c1; sC[2] += c2;
        qC += 510.f * c1 * c1 + c0 * c0 + c2 * c2;
    }
    __shared__ float red[8][256];
    red[0][t] = sR[0]; red[1][t] = sR[1]; red[2][t] = sR[2]; red[3][t] = qR;
    red[4][t] = sC[0]; red[5][t] = sC[1]; red[6][t] = sC[2]; red[7][t] = qC;
    __syncthreads();
    for (int off = 128; off > 0; off >>= 1) {
        if (t < off)
            for (int k = 0; k < 8; ++k) red[k][t] += red[k][t + off];
        __syncthreads();
    }
    if (t == 0) {
        float S_R[3], S_C[3];
        float rF[3], rL[3], cF[3], cL[3], rM[3], cM[3];
        for (int e2 = 0; e2 < 3; ++e2) {
            S_R[e2] = red[e2][0];
            S_C[e2] = red[4 + e2][0];
            rF[e2] = R[((b * 3 + e2) * L_ + 0) * NOUT + o];
            rL[e2] = R[((b * 3 + e2) * L_ + (L_ - 1)) * NOUT + o];
            cF[e2] = C[((b * 3 + e2) * L_ + 0) * NOUT + o];
            cL[e2] = C[((b * 3 + e2) * L_ + (L_ - 1)) * NOUT + o];
            rM[e2] = S_R[e2] - rF[e2] - rL[e2];
            cM[e2] = S_C[e2] - cF[e2] - cL[e2];
        }
        float cross =
            rF[0] * cF[0] + rF[1] * cM[0] + rF[2] * cL[0] +
            rM[0] * cF[1] + rM[1] * cM[1] + rM[2] * cL[1] +
            rL[0] * cF[2] + rL[1] * cM[2] + rL[2] * cL[2];
        float sum   = S_R[0] + 510.f * S_R[1] + S_R[2]
                    + S_C[0] + 510.f * S_C[1] + S_C[2];
        float sumsq = red[3][0] + red[7][0] + 2.f * cross;
        const float invN = 1.0f / (float)(L_ * L_);
        float mean = sum * invN;
        float var  = sumsq * invN - mean * mean;
        meanv[b * NOUT + o] = mean;
        invv[b * NOUT + o]  = rsqrtf(var + EPSV);
    }
}

// ---------------------------------------------------------------------------
// Fold mean/inv-std into R' = R*inv, C' = (C-mean)*inv so the combine pass is
// a single add per output element.
// ---------------------------------------------------------------------------
__global__ void fold(const float* __restrict__ R, const float* __restrict__ C,
                     const float* __restrict__ meanv, const float* __restrict__ invv,
                     float* __restrict__ Rp, float* __restrict__ Cp) {
    int idx = blockIdx.x * blockDim.x + threadIdx.x;
    if (idx >= B_ * 3 * L_ * NOUT) return;
    int o = idx % NOUT;
    int b = idx / (3 * L_ * NOUT);
    float iv = invv[b * NOUT + o];
    float mn = meanv[b * NOUT + o];
    Rp[idx] = R[idx] * iv;
    Cp[idx] = (C[idx] - mn) * iv;
}

// ---------------------------------------------------------------------------
// Combine: out[b,i,j,o] = R'[b,i,o,jc(j)] + C'[b,j,o,ic(i)].
// One block per (b,i) row; v4f coalesced, non-temporal stores of 201 MB.
// C' rows (~0.6 MB per batch) stay resident in the 192 MB L2.
// ---------------------------------------------------------------------------
__global__ __launch_bounds__(256)
void combine(const float* __restrict__ Rp, const float* __restrict__ Cp,
             float* __restrict__ out) {
    const int bi = blockIdx.x;
    const int b = bi / L_, i = bi % L_;
    const int t = threadIdx.x;
    const int ic = (i == 0) ? 0 : ((i == L_ - 1) ? 2 : 1);
    const v4f* Crow = (const v4f*)(Cp + ((size_t)(b * 3 + ic)) * L_ * NOUT);
    const float* R0 = Rp + (((size_t)b * 3 + 0) * L_ + i) * NOUT;
    const float* R1 = Rp + (((size_t)b * 3 + 1) * L_ + i) * NOUT;
    const float* R2 = Rp + (((size_t)b * 3 + 2) * L_ + i) * NOUT;
    v4f* orow = (v4f*)(out + ((size_t)(b * L_ + i)) * L_ * NOUT);
    const int nvec = L_ * NOUT / 4;                          // 12288
    for (int f = t; f < nvec; f += 256) {
        const int flat = f * 4;
        const int j = flat / NOUT;
        const int o = flat - j * NOUT;                       // o % 4 == 0
        const float* Rrow = (j == 0) ? R0 : ((j == L_ - 1) ? R2 : R1);
        v4f rv = *(const v4f*)(Rrow + o);
        v4f cv = Crow[f];
        v4f ov = rv + cv;
        __builtin_nontemporal_store(ov, &orow[f]);
    }
}

// ---------------------------------------------------------------------------
extern "C" void kernel_launch(void* const* d_in, const int* in_sizes, int n_in,
                              void* d_out, int out_size, void* d_ws, size_t ws_size,
                              hipStream_t stream) {
    const float* input1 = (const float*)d_in[0];   // [2,512,64]
    const float* input2 = (const float*)d_in[1];   // [2,512,64]
    const float* W      = (const float*)d_in[2];   // [96,128,3,3]
    float* out = (float*)d_out;                    // [2,512,512,96]

    // Workspace layout (floats)
    float* ws = (float*)d_ws;
    const size_t szW  = 3 * KK * NOUT;             // 55296
    const size_t szRC = (size_t)B_ * 3 * L_ * NOUT;// 294912
    float* Wr    = ws;                 ws += szW;
    float* Wc    = ws;                 ws += szW;
    float* R     = ws;                 ws += szRC;
    float* C     = ws;                 ws += szRC;
    float* Rp    = ws;                 ws += szRC;
    float* Cp    = ws;                 ws += szRC;
    float* meanv = ws;                 ws += B_ * NOUT;
    float* invv  = ws;

    // 1) collapse conv weights per edge class (pair-interleaved layout)
    {
        int n = 3 * KK * NOUT;
        prep_weights<<<(n + 255) / 256, 256, 0, stream>>>(W, Wr, Wc);
    }
    // 2) row-GEMM and col-GEMM via fp32 WMMA (one wave per 16x16 tile)
    {
        dim3 grid(L_ / 16, NOUT / 16, B_ * 3);
        gemm_rc<<<grid, 32, 0, stream>>>(input1, Wr, R);
        gemm_rc<<<grid, 32, 0, stream>>>(input2, Wc, C);
    }
    // 3) analytic InstanceNorm statistics
    stats<<<B_ * NOUT, 256, 0, stream>>>(R, C, meanv, invv);
    // 4) fold normalization into R', C'
    {
        int n = (int)(B_ * 3 * L_ * NOUT);
        fold<<<(n + 255) / 256, 256, 0, stream>>>(R, C, meanv, invv, Rp, Cp);
    }
    // 5) bandwidth-bound broadcast-add, NT stores
    combine<<<B_ * L_, 256, 0, stream>>>(Rp, Cp, out);
}